// TGATLayer_41961830482200
// MI455X (gfx1250) — compile-verified
//
#include <hip/hip_runtime.h>
#include <hip/hip_bf16.h>

typedef float v2f __attribute__((ext_vector_type(2)));
typedef float v8f __attribute__((ext_vector_type(8)));

#define NNODES 50000
#define NEDGES 800000
#define DIN    128
#define DOUT   128      // HEADS * C
#define HEADS  2

// ---- monotone float<->uint key for atomic max on fp32 ----
__device__ __forceinline__ unsigned f32key(float f) {
    unsigned b = __float_as_uint(f);
    return b ^ ((unsigned)((int)b >> 31) | 0x80000000u);
}
__device__ __forceinline__ float keyf32(unsigned k) {
    unsigned b = (k & 0x80000000u) ? (k & 0x7FFFFFFFu) : ~k;
    return __uint_as_float(b);
}

// ---------------- zero-init workspace ----------------
__global__ __launch_bounds__(256)
void zero_ws(float* __restrict__ p, long long n) {
    long long i = (long long)blockIdx.x * 256 + threadIdx.x;
    if (i < n) p[i] = 0.0f;
}

// ---------------- fused 4-way GEMM: o = x @ W + b, fp32 WMMA 16x16x4 ----
// grid = (N/16, 1, 4); block = 256 (8 waves); wave w owns 16-col tile w.
__global__ __launch_bounds__(256)
void gemm_qkvs(const float* __restrict__ x,
               const float* __restrict__ W0, const float* __restrict__ W1,
               const float* __restrict__ W2, const float* __restrict__ W3,
               const float* __restrict__ b0, const float* __restrict__ b1,
               const float* __restrict__ b2, const float* __restrict__ b3,
               float* __restrict__ o0, float* __restrict__ o1,
               float* __restrict__ o2, float* __restrict__ o3)
{
    const float* W; const float* bias; float* out;
    switch (blockIdx.z) {
        case 0:  W = W0; bias = b0; out = o0; break;
        case 1:  W = W1; bias = b1; out = o1; break;
        case 2:  W = W2; bias = b2; out = o2; break;
        default: W = W3; bias = b3; out = o3; break;
    }
    const int lane  = threadIdx.x & 31;
    const int wave  = threadIdx.x >> 5;
    const int m0    = blockIdx.x * 16;
    const int n0    = wave * 16;
    const int nlane = lane & 15;            // A: row M; B/C/D: col N
    const int khalf = (lane >> 4) * 2;      // K pair selector {0,1} vs {2,3}

    const float* arow = x + (size_t)(m0 + nlane) * DIN;
    v8f acc = {};
    #pragma unroll 4
    for (int k = 0; k < DIN; k += 4) {
        v2f a = *(const v2f*)(arow + k + khalf);               // A[M][k+khalf..+1]
        v2f b;
        b[0] = W[(size_t)(k + khalf)     * DOUT + n0 + nlane]; // B[k+khalf][N]
        b[1] = W[(size_t)(k + khalf + 1) * DOUT + n0 + nlane];
        acc = __builtin_amdgcn_wmma_f32_16x16x4_f32(
            /*neg_a=*/false, a, /*neg_b=*/false, b,
            /*c_mod=*/(short)0, acc, /*reuse_a=*/false, /*reuse_b=*/false);
    }
    const float bv = bias[n0 + nlane];
    const int rbase = m0 + (lane >> 4) * 8;   // D: row = vgpr + 8*lane[4]
    #pragma unroll
    for (int i = 0; i < 8; ++i)
        out[(size_t)(rbase + i) * DOUT + n0 + nlane] = acc[i] + bv;
}

// ---------------- per-edge logits + segment max (one wave / edge) ------
// lanes 0-15 -> head 0 (channels 0..63), lanes 16-31 -> head 1.
__global__ __launch_bounds__(256)
void edge_logits(const long long* __restrict__ ei,
                 const float* __restrict__ q, const float* __restrict__ k,
                 float* __restrict__ alpha, unsigned* __restrict__ mkey)
{
    const int lane = threadIdx.x & 31;
    const long long e = (long long)blockIdx.x * 8 + (threadIdx.x >> 5);
    const long long s = ei[e];
    const long long d = ei[(long long)NEDGES + e];

    float4 qv = ((const float4*)(q + d * DIN))[lane];
    float4 kv = ((const float4*)(k + s * DIN))[lane];
    float p = qv.x * kv.x + qv.y * kv.y + qv.z * kv.z + qv.w * kv.w;
    p += __shfl_xor(p, 1, 32);
    p += __shfl_xor(p, 2, 32);
    p += __shfl_xor(p, 4, 32);
    p += __shfl_xor(p, 8, 32);       // reduced within each 16-lane head group
    p *= 0.125f;                     // 1/sqrt(C), C = 64

    const int head = lane >> 4;
    if ((lane & 15) == 0) {
        alpha[e * HEADS + head] = p;
        atomicMax(&mkey[d * HEADS + head], f32key(p));
    }
}

// ---------------- exp, denom, weighted scatter-add of v ----------------
__global__ __launch_bounds__(256)
void edge_scatter(const long long* __restrict__ ei,
                  const float* __restrict__ v,
                  const float* __restrict__ alpha,
                  const unsigned* __restrict__ mkey,
                  float* __restrict__ denom,
                  float* __restrict__ agg)
{
    const int lane = threadIdx.x & 31;
    const long long e = (long long)blockIdx.x * 8 + (threadIdx.x >> 5);
    const long long s = ei[e];
    const long long d = ei[(long long)NEDGES + e];
    const int head = lane >> 4;

    const float al = alpha[e * HEADS + head];
    const float m  = keyf32(mkey[d * HEADS + head]);
    const float a  = __expf(al - m);
    if ((lane & 15) == 0) atomicAdd(&denom[d * HEADS + head], a);

    float4 vv = ((const float4*)(v + s * DIN))[lane];
    float* ag = agg + d * DIN + lane * 4;
    atomicAdd(ag + 0, a * vv.x);
    atomicAdd(ag + 1, a * vv.y);
    atomicAdd(ag + 2, a * vv.z);
    atomicAdd(ag + 3, a * vv.w);
}

// ---------------- finalize: out = agg/denom + skip (skip already in out)
__global__ __launch_bounds__(256)
void finalize(const float* __restrict__ agg, const float* __restrict__ denom,
              float* __restrict__ out)
{
    const long long i = (long long)blockIdx.x * 256 + threadIdx.x; // N*128
    const int node = (int)(i >> 7);
    const int head = ((int)i & 127) >> 6;
    float dn = fmaxf(denom[node * HEADS + head], 1e-16f);
    out[i] = out[i] + agg[i] / dn;
}

extern "C" void kernel_launch(void* const* d_in, const int* in_sizes, int n_in,
                              void* d_out, int out_size, void* d_ws, size_t ws_size,
                              hipStream_t stream) {
    const float*     x     = (const float*)d_in[0];
    const long long* ei    = (const long long*)d_in[1];   // [2, E] int64
    const float*     Wq    = (const float*)d_in[2];
    const float*     bq    = (const float*)d_in[3];
    const float*     Wk    = (const float*)d_in[4];
    const float*     bk    = (const float*)d_in[5];
    const float*     Wv    = (const float*)d_in[6];
    const float*     bv    = (const float*)d_in[7];
    const float*     Wsk   = (const float*)d_in[8];
    const float*     bsk   = (const float*)d_in[9];
    float* out = (float*)d_out;

    // workspace layout (fp32 elements)
    float*    agg   = (float*)d_ws;                                   // N*128
    float*    denom = agg + (size_t)NNODES * DIN;                     // N*2
    unsigned* mkey  = (unsigned*)(denom + (size_t)NNODES * HEADS);    // N*2
    float*    q     = (float*)(mkey + (size_t)NNODES * HEADS);        // N*128
    float*    k     = q + (size_t)NNODES * DIN;                       // N*128
    float*    v     = k + (size_t)NNODES * DIN;                       // N*128
    float*    alpha = v + (size_t)NNODES * DIN;                       // E*2

    // zero agg + denom + mkey (contiguous at start of ws)
    const long long nz = (long long)NNODES * DIN + (long long)NNODES * HEADS * 2;
    zero_ws<<<(unsigned)((nz + 255) / 256), 256, 0, stream>>>((float*)d_ws, nz);

    dim3 gg(NNODES / 16, 1, 4);    // 50000 = 3125 * 16, exact
    gemm_qkvs<<<gg, 256, 0, stream>>>(x, Wq, Wk, Wv, Wsk, bq, bk, bv, bsk,
                                      q, k, v, out /* skip -> d_out */);

    edge_logits <<<NEDGES / 8, 256, 0, stream>>>(ei, q, k, alpha, mkey);
    edge_scatter<<<NEDGES / 8, 256, 0, stream>>>(ei, v, alpha, mkey, denom, agg);
    finalize<<<(NNODES * DOUT) / 256, 256, 0, stream>>>(agg, denom, out);
}